// TokenChoiceGating_57475252355398
// MI455X (gfx1250) — compile-verified
//
#include <hip/hip_runtime.h>

// Problem constants (fixed by the reference)
#define B_        2
#define N_        4096
#define D_        512
#define E_        8
#define K_        2
#define CAP_      1024
#define EPS_      1e-9f
#define SINK_IT   6

typedef float v2f __attribute__((ext_vector_type(2)));
typedef float v4f __attribute__((ext_vector_type(4)));
typedef float v8f __attribute__((ext_vector_type(8)));

// ---------------------------------------------------------------------------
// Kernel 1: zero-fill the dense outputs (dispatch + combine + 2 scalar zeros).
// 512 MB of stores -> the roofline cost of the whole op (~22us @ 23.3 TB/s).
// b128 non-temporal stores: write-once stream, keep it out of L2's way.
// ---------------------------------------------------------------------------
__global__ __launch_bounds__(256) void zero_fill(float* __restrict__ out, size_t nfloat) {
    size_t n4 = nfloat >> 2;
    v4f* o4 = reinterpret_cast<v4f*>(out);
    size_t i = (size_t)blockIdx.x * blockDim.x + threadIdx.x;
    size_t stride = (size_t)gridDim.x * blockDim.x;
    v4f z = {};
    for (; i < n4; i += stride) __builtin_nontemporal_store(z, &o4[i]);
    // tail (the two scalar "zero" aux-loss outputs)
    if (blockIdx.x == 0 && threadIdx.x < (nfloat & 3)) out[(n4 << 2) + threadIdx.x] = 0.f;
}

// ---------------------------------------------------------------------------
// Kernel 2: gate logits = key_tokens[B*N, D] x query[E, D]^T, scaled by
// 1/exp(softmax_temp). V_WMMA_F32_16X16X4_F32: one wave computes a
// 16(rows) x 16(cols, experts padded 8->16) tile, looping K=512 in steps of 4.
//
// A 16x4 f32 layout (ISA 7.12.2): lanes 0-15 hold M=0..15 {K=k0,k0+1} in
// VGPR0/1, lanes 16-31 hold {K=k0+2,k0+3}.
// B 4x16 layout (rows striped across lanes per VGPR, half-lane row offset 2):
// VGPR0 = rows {k0, k0+2}, VGPR1 = rows {k0+1, k0+3}.
// C/D 16x16: VGPR r = row (r + half*8), col = lane&15.
//
// Branch-free: lanes with l>=8 load a duplicated (valid) query row and
// multiply by 0, so no EXEC manipulation appears anywhere near the WMMAs.
// ---------------------------------------------------------------------------
__global__ __launch_bounds__(256) void gate_logits_wmma(
    const float* __restrict__ keys,     // [B*N, D]
    const float* __restrict__ query,    // [E, D]
    const float* __restrict__ temp,     // [1] (log temperature)
    float* __restrict__ logits)         // [B*N, E] scaled logits
{
    const int wave = (int)((blockIdx.x * blockDim.x + threadIdx.x) >> 5);
    const int lane = threadIdx.x & 31;
    const int half = lane >> 4;     // 0: K sub-cols {0,1}; 1: {2,3}
    const int l    = lane & 15;     // A row (M) / B col (N)
    const int M0   = wave * 16;     // grid sized exactly: no bounds divergence

    const float* arow  = keys + (size_t)(M0 + l) * D_;
    const float* qrow  = query + (size_t)(l & 7) * D_;   // valid for every lane
    const float  bmask = (l < E_) ? 1.0f : 0.0f;         // zero the padded cols

    v8f acc = {};
#pragma unroll 4
    for (int k = 0; k < D_; k += 4) {
        v4f av = *reinterpret_cast<const v4f*>(arow + k);
        v4f bv = *reinterpret_cast<const v4f*>(qrow + k);
        v2f a; a.x = half ? av.z : av.x;           a.y = half ? av.w : av.y;
        v2f b; b.x = (half ? bv.z : bv.x) * bmask; b.y = (half ? bv.w : bv.y) * bmask;
        // D = A x B + C   (args: neg_a, A, neg_b, B, c_mod, C, reuse_a, reuse_b)
        acc = __builtin_amdgcn_wmma_f32_16x16x4_f32(false, a, false, b,
                                                    (short)0, acc, false, false);
    }

    const float scale = 1.0f / __expf(temp[0]);
    if (l < E_) {
#pragma unroll
        for (int r = 0; r < 8; ++r) {
            logits[(size_t)(M0 + r + half * 8) * E_ + l] = acc[r] * scale;
        }
    }
}

// ---------------------------------------------------------------------------
// Kernel 3: per-batch Sinkhorn (6 iters) + top-2 routing + ordered capacity
// assignment + sparse scatter into dispatch/combine.
// One 1024-thread block per b; each thread owns 4 consecutive tokens in
// registers (4x8 f32). LDS (64 KB) used only for tree reductions / scans.
// ---------------------------------------------------------------------------
__global__ __launch_bounds__(1024) void sinkhorn_route(
    const float* __restrict__ logits,   // [B, N, E] scaled logits
    const float* __restrict__ probs,    // [K, B, N]
    float* __restrict__ out)            // dispatch[B,N,E,CAP] ++ combine[...] ++ 2 zeros
{
    __shared__ float red_m[1024 * E_];  // 32 KB
    __shared__ float red_s[1024 * E_];  // 32 KB
    unsigned* scan = reinterpret_cast<unsigned*>(red_m);  // reused for the prefix scan

    const int b   = blockIdx.x;
    const int tid = threadIdx.x;
    const int n0  = tid * 4;            // first of 4 consecutive tokens

    // ---- load this thread's 4 tokens x 8 experts into registers ----
    float t[4][E_];
    {
        const float* src = logits + ((size_t)b * N_ + n0) * E_;
#pragma unroll
        for (int j = 0; j < 4; ++j)
#pragma unroll
            for (int e = 0; e < E_; ++e) t[j][e] = src[j * E_ + e];
    }

    // ---- Sinkhorn: 6 x (column logsumexp over N, then row logsumexp over E) ----
    for (int it = 0; it < SINK_IT; ++it) {
        // per-thread partial (max, sum-exp) per expert
        float m[E_], s[E_];
#pragma unroll
        for (int e = 0; e < E_; ++e) {
            float mm = fmaxf(fmaxf(t[0][e], t[1][e]), fmaxf(t[2][e], t[3][e]));
            float ss = 0.f;
#pragma unroll
            for (int j = 0; j < 4; ++j) ss += __expf(t[j][e] - mm);
            m[e] = mm; s[e] = ss;
        }
#pragma unroll
        for (int e = 0; e < E_; ++e) { red_m[tid * E_ + e] = m[e]; red_s[tid * E_ + e] = s[e]; }
        __syncthreads();
        // tree combine of (max, scaled-sum) pairs -> totals at thread 0
        for (int str = 512; str >= 1; str >>= 1) {
            if (tid < str) {
#pragma unroll
                for (int e = 0; e < E_; ++e) {
                    float m1 = red_m[tid * E_ + e],         s1 = red_s[tid * E_ + e];
                    float m2 = red_m[(tid + str) * E_ + e], s2 = red_s[(tid + str) * E_ + e];
                    float mm = fmaxf(m1, m2);
                    red_s[tid * E_ + e] = s1 * __expf(m1 - mm) + s2 * __expf(m2 - mm);
                    red_m[tid * E_ + e] = mm;
                }
            }
            __syncthreads();
        }
        float col[E_];
#pragma unroll
        for (int e = 0; e < E_; ++e) col[e] = red_m[e] + __logf(red_s[e]);
        __syncthreads();  // protect LDS before next iteration's writes
#pragma unroll
        for (int j = 0; j < 4; ++j)
#pragma unroll
            for (int e = 0; e < E_; ++e) t[j][e] -= col[e];

        // row logsumexp (E=8, thread-local)
#pragma unroll
        for (int j = 0; j < 4; ++j) {
            float mm = t[j][0];
#pragma unroll
            for (int e = 1; e < E_; ++e) mm = fmaxf(mm, t[j][e]);
            float ss = 0.f;
#pragma unroll
            for (int e = 0; e < E_; ++e) ss += __expf(t[j][e] - mm);
            float lse = mm + __logf(ss);
#pragma unroll
            for (int e = 0; e < E_; ++e) t[j][e] -= lse;
        }
    }

    // ---- raw gates, top-2, normalization, stochastic routing ----
    float g[4][E_];
#pragma unroll
    for (int j = 0; j < 4; ++j)
#pragma unroll
        for (int e = 0; e < E_; ++e) g[j][e] = __expf(t[j][e]);

    int i0[4], i1[4];
    float g0[4], g1[4];
    bool act1[4];
#pragma unroll
    for (int j = 0; j < 4; ++j) {
        int b0 = 0; float v0 = g[j][0];
#pragma unroll
        for (int e = 1; e < E_; ++e) if (g[j][e] > v0) { v0 = g[j][e]; b0 = e; }
        int b1 = 0; float v1 = -1.0f;
#pragma unroll
        for (int e = 0; e < E_; ++e) if (e != b0 && g[j][e] > v1) { v1 = g[j][e]; b1 = e; }
        float denom = fmaxf(v0 + v1, EPS_);
        i0[j] = b0; i1[j] = b1;
        g0[j] = v0 / denom; g1[j] = v1 / denom;
        // k=0 is forced True; k=1: probs[1,b,n] < g1 / 0.2
        float p = probs[(size_t)1 * B_ * N_ + (size_t)b * N_ + (n0 + j)];
        act1[j] = p < (g1[j] * 5.0f);
    }

    const size_t COMBINE_OFF = (size_t)B_ * N_ * E_ * CAP_;
    unsigned prev[E_];
#pragma unroll
    for (int e = 0; e < E_; ++e) prev[e] = 0;

    // ---- K=2 sequential passes: ordered cumsum over tokens, capacity, scatter ----
    for (int kk = 0; kk < K_; ++kk) {
        unsigned cnt[E_];
#pragma unroll
        for (int e = 0; e < E_; ++e) cnt[e] = 0;

        int eidx[4]; float gv[4]; bool act[4]; unsigned lpos[4];
#pragma unroll
        for (int j = 0; j < 4; ++j) {
            eidx[j] = (kk == 0) ? i0[j] : i1[j];
            gv[j]   = (kk == 0) ? g0[j] : g1[j];
            act[j]  = (kk == 0) ? true  : act1[j];
            lpos[j] = 0;
            if (act[j]) { lpos[j] = cnt[eidx[j]]; cnt[eidx[j]] += 1u; }
        }

        // inclusive Hillis-Steele scan of 8-expert count vectors across threads
#pragma unroll
        for (int e = 0; e < E_; ++e) scan[tid * E_ + e] = cnt[e];
        __syncthreads();
        for (int str = 1; str < 1024; str <<= 1) {
            unsigned v[E_];
            if (tid >= str) {
#pragma unroll
                for (int e = 0; e < E_; ++e) v[e] = scan[(tid - str) * E_ + e];
            }
            __syncthreads();
            if (tid >= str) {
#pragma unroll
                for (int e = 0; e < E_; ++e) scan[tid * E_ + e] += v[e];
            }
            __syncthreads();
        }
        unsigned excl[E_], tot[E_];
#pragma unroll
        for (int e = 0; e < E_; ++e) {
            excl[e] = scan[tid * E_ + e] - cnt[e];       // exclusive prefix for this thread
            tot[e]  = scan[1023 * E_ + e];               // total mask count this pass
        }
        __syncthreads();  // scan buffer reused next pass

        // positions + sparse scatter (dispatch=1.0, combine=gate) when in capacity
#pragma unroll
        for (int j = 0; j < 4; ++j) {
            if (act[j]) {
                unsigned pos = prev[eidx[j]] + excl[eidx[j]] + lpos[j];
                if (pos < (unsigned)CAP_) {
                    size_t idx = (((size_t)b * N_ + (n0 + j)) * E_ + eidx[j]) * CAP_ + pos;
                    out[idx] = 1.0f;                    // dispatch
                    out[COMBINE_OFF + idx] = gv[j];     // combine
                }
            }
        }
        // prev_expert_count += post-capacity kept count == min(prev+tot, CAP)
#pragma unroll
        for (int e = 0; e < E_; ++e) {
            unsigned p2 = prev[e] + tot[e];
            prev[e] = (p2 > (unsigned)CAP_) ? (unsigned)CAP_ : p2;
        }
    }
}

// ---------------------------------------------------------------------------
// Launch: d_in order = {key_tokens[B,N,D], query[E,D], softmax_temp[1], probs[K,B,N]}
// d_out  = dispatch[B,N,E,CAP] ++ combine[B,N,E,CAP] ++ zero ++ zero (f32)
// ---------------------------------------------------------------------------
extern "C" void kernel_launch(void* const* d_in, const int* in_sizes, int n_in,
                              void* d_out, int out_size, void* d_ws, size_t ws_size,
                              hipStream_t stream) {
    const float* keys  = (const float*)d_in[0];
    const float* query = (const float*)d_in[1];
    const float* temp  = (const float*)d_in[2];
    const float* probs = (const float*)d_in[3];
    float* out    = (float*)d_out;
    float* logits = (float*)d_ws;   // [B*N, E] = 256 KB scratch

    const size_t total_floats = (size_t)2 * B_ * N_ * E_ * CAP_ + 2;  // 134,217,730

    // 1) zero the dense outputs (dominant cost; ~512 MB of streaming stores)
    zero_fill<<<4096, 256, 0, stream>>>(out, total_floats);

    // 2) gate logits via WMMA: 512 tiles of 16 rows, 1 wave/tile, 8 waves/block
    gate_logits_wmma<<<(B_ * N_ / 16) / 8, 256, 0, stream>>>(keys, query, temp, logits);

    // 3) Sinkhorn + routing + sparse scatter: one 1024-thread block per batch
    sinkhorn_route<<<B_, 1024, 0, stream>>>(logits, probs, out);
}